// PairwiseMaxMarginHingeLossFunc_21775484190860
// MI455X (gfx1250) — compile-verified
//
#include <hip/hip_runtime.h>

// CDNA5 / gfx1250: wave32, WMMA F32 16x16x4 (A,B = v2f per lane; C/D = v8f per lane)
typedef __attribute__((ext_vector_type(2))) float v2f;
typedef __attribute__((ext_vector_type(8))) float v8f;

#define N_LIST 1024                       // list length (pairwise matrix N x N)
#define BATCH 32                          // batch size
#define TILE 16                           // WMMA tile edge
#define KTILES (N_LIST / TILE)            // 64 column tiles
#define WAVES_PER_BLOCK 8
#define THREADS (WAVES_PER_BLOCK * 32)    // 256 threads = 8 wave32
#define ITILE_GROUPS (KTILES / WAVES_PER_BLOCK)  // 8 row-tile groups per batch
#define NBLOCKS (BATCH * ITILE_GROUPS)    // 256 workgroups

// relu: clamp to [0, inf). Lowered by clang to dual-issued max pair; at the
// epilogue's VALU-slot floor either way (see v_dual_max_num + v_pk_add_f32).
__device__ __forceinline__ float relu1(float x) {
    return __builtin_amdgcn_fmed3f(x, 0.0f, __builtin_inff());
}

// Rank-4 factorization:  M[j,k] = (o_j-o_k)(l_j-l_k) = u_j + u_k - o_j l_k - l_j o_k
//   A (16x4)  rows j: [ u_j, 1, o_j, l_j ]
//   B'(4x16)  cols k: [ -1, -u_k, l_k, o_k ]^T      (negated so A*B' = -M)
//   D = A*B' + C,  C = margin  =>  D = margin - M   (exactly the hinge argument)
//
// B' operand pairs are precomputed in LDS in the exact per-lane WMMA layout
// (half 0 = K{0,1}, half 1 = K{2,3}), so the inner loop is just:
//   ds_load_b64 -> v_wmma_f32_16x16x4_f32 -> relu+pk_add epilogue.
__global__ __launch_bounds__(THREADS)
void pairwise_hinge_wmma_kernel(const float* __restrict__ gout,
                                const float* __restrict__ glab,
                                const float* __restrict__ gmargin,
                                float* __restrict__ partial)
{
    __shared__ float o_s[N_LIST];
    __shared__ float l_s[N_LIST];
    __shared__ v2f   bprep[2][N_LIST];    // [half][n]: ready-to-use B' lane pairs
    __shared__ float wsum[WAVES_PER_BLOCK];

    const int tid = threadIdx.x;
    const int b   = blockIdx.x / ITILE_GROUPS;   // batch column
    const int ig  = blockIdx.x % ITILE_GROUPS;   // row-tile group

    // Stage this batch's columns o[:,b], l[:,b] (layout is [N, B]) and build
    // the B' operand table once.
    for (int n = tid; n < N_LIST; n += THREADS) {
        const float o = gout[n * BATCH + b];
        const float l = glab[n * BATCH + b];
        o_s[n] = o;
        l_s[n] = l;
        const float u = o * l;
        v2f b0; b0.x = -1.0f; b0.y = -u;   // K=0: -1,  K=1: -u_k
        v2f b1; b1.x = l;     b1.y = o;    // K=2: l_k, K=3: o_k
        bprep[0][n] = b0;
        bprep[1][n] = b1;
    }
    __syncthreads();

    const float margin = gmargin[0];

    const int lane = tid & 31;
    const int wave = tid >> 5;
    const int m    = lane & 15;   // row/col within tile
    const int half = lane >> 4;   // K-pair select (lanes 0-15: K=0,1; 16-31: K=2,3)

    // Loop-invariant A operand for this wave's 16-row strip.
    const int i0 = (ig * WAVES_PER_BLOCK + wave) * TILE;
    const float om = o_s[i0 + m];
    const float lm = l_s[i0 + m];
    v2f a;
    a.x = half ? om : om * lm;   // K=0: u_j      K=2: o_j
    a.y = half ? lm : 1.0f;      // K=1: 1        K=3: l_j

    // This lane's B' stream: fixed base, 16-element (128 B) stride -> DS
    // immediate offsets.
    const v2f* __restrict__ bptr = &bprep[half][m];

    // 8 independent accumulator chains -> v_pk_add_f32 pairs, no serial chain.
    v8f accv = { 0.f, 0.f, 0.f, 0.f, 0.f, 0.f, 0.f, 0.f };

#pragma unroll 4
    for (int kt = 0; kt < KTILES; ++kt) {
        const v2f bb = bptr[kt * TILE];   // ds_load_b64, zero VALU operand prep

        v8f c = { margin, margin, margin, margin,
                  margin, margin, margin, margin };
        // D = margin - M  for the full 16x16 tile in one matrix op.
        v8f d = __builtin_amdgcn_wmma_f32_16x16x4_f32(
            /*neg_a=*/false, a, /*neg_b=*/false, bb,
            /*c_mod=*/(short)0, c, /*reuse_a=*/false, /*reuse_b=*/false);

#pragma unroll
        for (int e = 0; e < 8; ++e)
            accv[e] += relu1(d[e]);
    }

    // Horizontal sum of the 8 accumulator slots, then wave32 reduction.
    float acc = (accv[0] + accv[1]) + (accv[2] + accv[3])
              + ((accv[4] + accv[5]) + (accv[6] + accv[7]));
#pragma unroll
    for (int off = 16; off > 0; off >>= 1)
        acc += __shfl_xor(acc, off, 32);
    if (lane == 0) wsum[wave] = acc;
    __syncthreads();

    if (tid == 0) {
        float s = 0.0f;
#pragma unroll
        for (int w = 0; w < WAVES_PER_BLOCK; ++w) s += wsum[w];
        partial[blockIdx.x] = s;
    }
}

// Deterministic final reduction of the 256 per-block partials + 1/(2N) scale.
__global__ __launch_bounds__(NBLOCKS)
void pairwise_hinge_reduce_kernel(const float* __restrict__ partial,
                                  float* __restrict__ out)
{
    __shared__ float s[NBLOCKS];
    const int tid = threadIdx.x;
    s[tid] = partial[tid];
    __syncthreads();
#pragma unroll
    for (int off = NBLOCKS / 2; off > 0; off >>= 1) {
        if (tid < off) s[tid] += s[tid + off];
        __syncthreads();
    }
    if (tid == 0) out[0] = s[0] * (1.0f / (2.0f * (float)N_LIST));
}

extern "C" void kernel_launch(void* const* d_in, const int* in_sizes, int n_in,
                              void* d_out, int out_size, void* d_ws, size_t ws_size,
                              hipStream_t stream)
{
    const float* gout    = (const float*)d_in[0];   // out   [N, B] f32
    const float* glab    = (const float*)d_in[1];   // label [N, B] f32
    const float* gmargin = (const float*)d_in[2];   // margin [1]  f32
    float* partial = (float*)d_ws;                  // 256 floats of scratch
    float* result  = (float*)d_out;                 // scalar loss

    pairwise_hinge_wmma_kernel<<<NBLOCKS, THREADS, 0, stream>>>(gout, glab, gmargin, partial);
    pairwise_hinge_reduce_kernel<<<1, NBLOCKS, 0, stream>>>(partial, result);
}